// LabelSmoothLoss_63926293233763
// MI455X (gfx1250) — compile-verified
//
#include <hip/hip_runtime.h>

typedef float v2f __attribute__((ext_vector_type(2)));
typedef float v8f __attribute__((ext_vector_type(8)));

#define NCLS   81      // classes (80 + background)
#define CPAD   85      // padded row pitch in LDS (odd -> bank-friendly)
#define RPB    128     // rows per block (8 waves * 16 rows)
#define TPB    256
#define KSTEPS 21      // ceil(81 / 4) WMMA K-steps

// Kernel 1: per-block partial losses.
// Each wave (32 lanes) owns 16 rows. dot(T[label], x) for the 16 rows is the
// diagonal of G x X^T computed with V_WMMA_F32_16X16X4_F32; max/LSE are
// 2-lanes-per-row LDS scans combined with shfl_xor.
__global__ __launch_bounds__(TPB) void lsl_main(const float* __restrict__ X,
                                                const float* __restrict__ Tbl,
                                                const int*   __restrict__ label,
                                                float*       __restrict__ partial)
{
    __shared__ float tile[RPB * CPAD];
    __shared__ float wsum[TPB / 32];

    const int tid     = threadIdx.x;
    const int lane    = tid & 31;
    const int wave    = tid >> 5;
    const int rowBase = blockIdx.x * RPB;

    // ---- stream tile in with b128 global loads, stage into LDS ----
    {
        const float4* __restrict__ Xv =
            reinterpret_cast<const float4*>(X) + (size_t)rowBase * NCLS / 4;
        const int nvec = RPB * NCLS / 4; // 2592 float4 per block
        for (int i = tid; i < nvec; i += TPB) {
            float4 v = Xv[i];
            int e = i * 4;
            #pragma unroll
            for (int j = 0; j < 4; ++j) {
                int ee = e + j;
                int r  = ee / NCLS;
                int c  = ee - r * NCLS;
                tile[r * CPAD + c] = (&v.x)[j];
            }
        }
        // zero pad columns 81..84 (WMMA K-padding reads up to col 83)
        for (int i = tid; i < RPB * 4; i += TPB) {
            int r = i >> 2;
            tile[r * CPAD + NCLS + (i & 3)] = 0.0f;
        }
    }
    __syncthreads();

    const int wr0  = wave * 16;           // wave's local row base in tile
    const int mA   = lane & 15;           // A/B row index within 16-row tile
    const int koff = (lane >> 4) << 1;    // lanes 0-15: K+{0,1}; lanes 16-31: K+{2,3}
    const int lbl  = label[rowBase + wr0 + mA];
    const int tRow = lbl * NCLS;
    const float* __restrict__ bRow = &tile[(wr0 + mA) * CPAD];

    v8f   acc  = {0.f, 0.f, 0.f, 0.f, 0.f, 0.f, 0.f, 0.f};
    float tsum = 0.0f;                    // per-lane sum of T entries (for S(l))

    #pragma unroll
    for (int s = 0; s < KSTEPS; ++s) {
        int   k0  = s * 4 + koff;
        int   k1  = k0 + 1;
        float m0  = (k0 < NCLS) ? 1.0f : 0.0f;   // branchless K-pad masks
        float m1  = (k1 < NCLS) ? 1.0f : 0.0f;
        int   ck0 = (k0 < NCLS) ? k0 : NCLS - 1;
        int   ck1 = (k1 < NCLS) ? k1 : NCLS - 1;
        v2f a, b;
        a.x = Tbl[tRow + ck0] * m0;   // gathered table rows (cached, 26 KB)
        a.y = Tbl[tRow + ck1] * m1;
        b.x = bRow[k0];               // LDS; pad cols are zeroed
        b.y = bRow[k1];
        acc = __builtin_amdgcn_wmma_f32_16x16x4_f32(
                  false, a, false, b, (short)0, acc, false, false);
        tsum += a.x + a.y;
    }

    // S(l) = full table-row sum: lanes l and l^16 hold complementary K halves
    float S = tsum + __shfl_xor(tsum, 16, 32);

    // Diagonal of D: row r<8 -> lane r, VGPR r; row r>=8 -> lane r+16, VGPR r-8
    float diag = 0.0f;
    int   di   = lane & 7;
    #pragma unroll
    for (int j = 0; j < 8; ++j)
        if (di == j) diag = acc[j];

    // ---- per-row max / logsumexp: 2 lanes per row, combined via shfl_xor(8)
    // lanes 0-7: rows 0-7 half0; 8-15: rows 0-7 half1;
    // lanes 16-23: rows 8-15 half0; 24-31: rows 8-15 half1.
    int half = (lane >> 3) & 1;
    int rr   = wr0 + (lane & 7) + ((lane >> 4) << 3);
    const float* __restrict__ rRow = &tile[rr * CPAD];

    float mx = -3.402823466e+38f;
    for (int c = half; c < NCLS; c += 2) mx = fmaxf(mx, rRow[c]);
    mx = fmaxf(mx, __shfl_xor(mx, 8, 32));

    float se = 0.0f;
    for (int c = half; c < NCLS; c += 2) se += __expf(rRow[c] - mx);
    se += __shfl_xor(se, 8, 32);
    float lse = __logf(se);

    // owner lanes carry both the diagonal and this row's S / max / lse
    bool  owner = (lane < 8) || (lane >= 24);
    float loss  = owner ? (S * (mx + lse) - diag) : 0.0f;

    // deterministic wave butterfly sum
    #pragma unroll
    for (int off = 16; off > 0; off >>= 1)
        loss += __shfl_xor(loss, off, 32);

    if (lane == 0) wsum[wave] = loss;
    __syncthreads();
    if (tid == 0) {
        float t = 0.0f;
        #pragma unroll
        for (int w = 0; w < TPB / 32; ++w) t += wsum[w];
        partial[blockIdx.x] = t;
    }
}

// Kernel 2: deterministic fixed-order reduction of per-block partials.
__global__ __launch_bounds__(256) void lsl_reduce(const float* __restrict__ partial,
                                                  int nparts, float scale,
                                                  float* __restrict__ out)
{
    __shared__ float s[256];
    float v = 0.0f;
    for (int i = threadIdx.x; i < nparts; i += 256) v += partial[i];
    s[threadIdx.x] = v;
    __syncthreads();
    for (int off = 128; off > 0; off >>= 1) {
        if ((int)threadIdx.x < off) s[threadIdx.x] += s[threadIdx.x + off];
        __syncthreads();
    }
    if (threadIdx.x == 0) out[0] = s[0] * scale;
}

extern "C" void kernel_launch(void* const* d_in, const int* in_sizes, int n_in,
                              void* d_out, int out_size, void* d_ws, size_t ws_size,
                              hipStream_t stream)
{
    const float* X     = (const float*)d_in[0];   // [N, 81] f32
    const float* Tbl   = (const float*)d_in[1];   // [81, 81] f32
    const int*   label = (const int*)d_in[2];     // [N] int
    float*       out   = (float*)d_out;           // scalar f32
    float*       part  = (float*)d_ws;            // nblocks floats of scratch

    const int N       = in_sizes[2];              // 262144
    const int nblocks = N / RPB;                  // 2048

    lsl_main  <<<nblocks, TPB, 0, stream>>>(X, Tbl, label, part);
    lsl_reduce<<<1,       256, 0, stream>>>(part, nblocks, 1.0f / (float)N, out);
}